// ResidualBlockWithChannel_26121991094584
// MI455X (gfx1250) — compile-verified
//
#include <hip/hip_runtime.h>
#include <hip/hip_bf16.h>

typedef __attribute__((ext_vector_type(2))) float v2f;
typedef __attribute__((ext_vector_type(8))) float v8f;

#define N_BATCH 2048
#define C_CH    64
#define F_DIM   256
#define CF      (C_CH * F_DIM)
#define BN_EPS  1e-3f

// block tile: 128 rows (M) x 64 cols (N-out), K stepped by 16
// 8 waves in 4x2; each wave computes a 32x32 register tile (4 accumulators)
#define BM 128
#define BN 64
#define BK 16
#define LDA (BK + 4)   // 20: conflict-free frag gathers + 16B-aligned b128 stores
#define LDB (BN + 16)  // 80: lane-halves land in disjoint bank sets

// --------------------------------------------------------------------------
// BN statistics over the batch dim, 4 features per thread (float4 streams),
// folded into scale/shift so the GEMM applies BN as a single FMA.
// --------------------------------------------------------------------------
__global__ __launch_bounds__(256)
void bn_stats_kernel(const float* __restrict__ src,
                     const float* __restrict__ gamma,
                     const float* __restrict__ beta,
                     float* __restrict__ scale,
                     float* __restrict__ shift)
{
    int j4 = (blockIdx.x * 256 + threadIdx.x) * 4;  // feature base in [0, C*F)
    if (j4 >= CF) return;
    float s0 = 0.f, s1 = 0.f, s2 = 0.f, s3 = 0.f;
    float q0 = 0.f, q1 = 0.f, q2 = 0.f, q3 = 0.f;
    const float* p = src + j4;
    #pragma unroll 4
    for (int n = 0; n < N_BATCH; ++n) {
        float4 v = *(const float4*)(p + (size_t)n * CF);
        s0 += v.x; q0 += v.x * v.x;
        s1 += v.y; q1 += v.y * v.y;
        s2 += v.z; q2 += v.z * v.z;
        s3 += v.w; q3 += v.w * v.w;
    }
    const float inv_n = 1.0f / (float)N_BATCH;
    float4 g = *(const float4*)(gamma + j4);
    float4 b = *(const float4*)(beta  + j4);
    float m0 = s0 * inv_n, m1 = s1 * inv_n, m2 = s2 * inv_n, m3 = s3 * inv_n;
    float4 sc, sh;
    sc.x = g.x * rsqrtf(q0 * inv_n - m0 * m0 + BN_EPS);
    sc.y = g.y * rsqrtf(q1 * inv_n - m1 * m1 + BN_EPS);
    sc.z = g.z * rsqrtf(q2 * inv_n - m2 * m2 + BN_EPS);
    sc.w = g.w * rsqrtf(q3 * inv_n - m3 * m3 + BN_EPS);
    sh.x = b.x - m0 * sc.x;
    sh.y = b.y - m1 * sc.y;
    sh.z = b.z - m2 * sc.z;
    sh.w = b.w - m3 * sc.w;
    *(float4*)(scale + j4) = sc;
    *(float4*)(shift + j4) = sh;
}

// --------------------------------------------------------------------------
// Fused  relu(src*scale+shift) @ w[c]^T + bias  (+ residual)
// Each wave: 2x2 grid of 16x16 f32 WMMA tiles -> 2 LDS b32 reads per WMMA.
// --------------------------------------------------------------------------
template <bool ADD_RES>
__global__ __launch_bounds__(256)
void chan_gemm_kernel(const float* __restrict__ src,    // (N, C, F)
                      const float* __restrict__ scale,  // (C*F)
                      const float* __restrict__ shift,  // (C*F)
                      const float* __restrict__ w,      // (C, F_out, F_in)
                      const float* __restrict__ bias,   // (C, F)
                      const float* __restrict__ resid,  // (N, C, F) or null
                      float* __restrict__ dst)          // (N, C, F)
{
    __shared__ float As[BM][LDA];   // 128x20 f32 = 10.0 KB
    __shared__ float Bs[BK][LDB];   //  16x80 f32 =  5.0 KB

    const int c  = blockIdx.z;
    const int m0 = blockIdx.y * BM;    // batch-row base
    const int o0 = blockIdx.x * BN;    // output-feature base
    const int cF = c * F_DIM;

    const int tid  = threadIdx.x;
    const int lane = tid & 31;
    const int wv   = tid >> 5;             // wave 0..7
    const int wmR  = (wv & 3) * 32;        // wave row-tile (4 groups of 32)
    const int wnC  = (wv >> 2) * 32;       // wave col-tile (2 groups of 32)
    const int l15  = lane & 15;
    const bool hi  = lane >= 16;

    const float* wc = w     + (size_t)c * F_DIM * F_DIM;
    const float* sc = scale + cF;
    const float* sh = shift + cF;

    // accumulators initialized with bias (bias depends only on output column)
    float bval[2];
    #pragma unroll
    for (int ni = 0; ni < 2; ++ni)
        bval[ni] = bias[cF + o0 + wnC + ni * 16 + l15];
    v8f acc[2][2];
    #pragma unroll
    for (int mi = 0; mi < 2; ++mi)
        #pragma unroll
        for (int ni = 0; ni < 2; ++ni)
            #pragma unroll
            for (int i = 0; i < 8; ++i) acc[mi][ni][i] = bval[ni];

    // static fragment addressing (ISA 7.12.2: 16x4 f32 A, 4x16 B)
    const int aK = hi ? 2 : 0;    // lanes 16-31 hold K=2,3
    const int bK = hi ? 1 : 0;    // B VGPR0: K0|K1 halves, VGPR1: K2|K3

    // staging coordinates
    const int aR0 = tid >> 2;              // 0..63 (+64 second pass)
    const int aKq = (tid & 3) * 4;         // 0,4,8,12
    const int bO  = tid >> 2;              // 0..63
    const int bKq = (tid & 3) * 4;

    for (int kk = 0; kk < F_DIM; kk += BK) {
        // stage A tile (128x16): BN+ReLU applied, 2 float4 per thread,
        // 16B-aligned global loads, b128 LDS stores (LDA keeps alignment)
        float4 sc4 = *(const float4*)(sc + kk + aKq);
        float4 sh4 = *(const float4*)(sh + kk + aKq);
        #pragma unroll
        for (int i = 0; i < 2; ++i) {
            int r = aR0 + i * 64;
            float4 v = *(const float4*)(src + (size_t)(m0 + r) * CF + cF + kk + aKq);
            float4 t;
            t.x = fmaxf(fmaf(v.x, sc4.x, sh4.x), 0.0f);
            t.y = fmaxf(fmaf(v.y, sc4.y, sh4.y), 0.0f);
            t.z = fmaxf(fmaf(v.z, sc4.z, sh4.z), 0.0f);
            t.w = fmaxf(fmaf(v.w, sc4.w, sh4.w), 0.0f);
            *(float4*)&As[r][aKq] = t;
        }
        // stage B tile (16x64) = transposed slice of w[c]: one 16B load per
        // thread along contiguous f(=k), register transpose, 4 b32 LDS stores
        {
            float4 v = *(const float4*)(wc + (size_t)(o0 + bO) * F_DIM + kk + bKq);
            Bs[bKq + 0][bO] = v.x;
            Bs[bKq + 1][bO] = v.y;
            Bs[bKq + 2][bO] = v.z;
            Bs[bKq + 3][bO] = v.w;
        }
        __syncthreads();

        #pragma unroll
        for (int k4 = 0; k4 < BK; k4 += 4) {
            v2f a[2], b[2];
            #pragma unroll
            for (int mi = 0; mi < 2; ++mi) {
                const int aRow = wmR + mi * 16 + l15;
                a[mi].x = As[aRow][k4 + aK + 0];
                a[mi].y = As[aRow][k4 + aK + 1];
            }
            #pragma unroll
            for (int ni = 0; ni < 2; ++ni) {
                const int bCol = wnC + ni * 16 + l15;
                b[ni].x = Bs[k4 + bK + 0][bCol];
                b[ni].y = Bs[k4 + bK + 2][bCol];
            }
            #pragma unroll
            for (int mi = 0; mi < 2; ++mi)
                #pragma unroll
                for (int ni = 0; ni < 2; ++ni)
                    acc[mi][ni] = __builtin_amdgcn_wmma_f32_16x16x4_f32(
                        /*neg_a=*/false, a[mi], /*neg_b=*/false, b[ni],
                        /*c_mod=*/(short)0, acc[mi][ni],
                        /*reuse_a=*/false, /*reuse_b=*/false);
        }
        __syncthreads();
    }

    // store D; C/D layout: VGPR r -> row r (lanes 0-15) / row r+8 (lanes 16-31)
    const int rHalf = hi ? 8 : 0;
    #pragma unroll
    for (int mi = 0; mi < 2; ++mi) {
        const int rBase = m0 + wmR + mi * 16 + rHalf;
        #pragma unroll
        for (int ni = 0; ni < 2; ++ni) {
            const int col = cF + o0 + wnC + ni * 16 + l15;
            #pragma unroll
            for (int r = 0; r < 8; ++r) {
                size_t off = (size_t)(rBase + r) * CF + col;
                float v = acc[mi][ni][r];
                if (ADD_RES) v += resid[off];
                dst[off] = v;
            }
        }
    }
}

// --------------------------------------------------------------------------
extern "C" void kernel_launch(void* const* d_in, const int* in_sizes, int n_in,
                              void* d_out, int out_size, void* d_ws, size_t ws_size,
                              hipStream_t stream)
{
    const float* x   = (const float*)d_in[0];
    const float* w0  = (const float*)d_in[1];
    const float* b0  = (const float*)d_in[2];
    const float* w1  = (const float*)d_in[3];
    const float* b1  = (const float*)d_in[4];
    const float* g0  = (const float*)d_in[5];
    const float* be0 = (const float*)d_in[6];
    const float* g1  = (const float*)d_in[7];
    const float* be1 = (const float*)d_in[8];
    float* out = (float*)d_out;

    // workspace: y0 (N*C*F f32 = 128 MiB) then 4 small stat vectors
    float* y0     = (float*)d_ws;
    float* scale0 = y0 + (size_t)N_BATCH * CF;
    float* shift0 = scale0 + CF;
    float* scale1 = shift0 + CF;
    float* shift1 = scale1 + CF;

    dim3 sgrid(CF / 4 / 256), sblock(256);
    dim3 ggrid(F_DIM / BN, N_BATCH / BM, C_CH), gblock(256);

    // layer 0: BN stats -> fused BN+ReLU+GEMM(w0)+bias
    bn_stats_kernel<<<sgrid, sblock, 0, stream>>>(x, g0, be0, scale0, shift0);
    chan_gemm_kernel<false><<<ggrid, gblock, 0, stream>>>(
        x, scale0, shift0, w0, b0, nullptr, y0);

    // layer 1: BN stats -> fused BN+ReLU+GEMM(w1)+bias + residual
    bn_stats_kernel<<<sgrid, sblock, 0, stream>>>(y0, g1, be1, scale1, shift1);
    chan_gemm_kernel<true><<<ggrid, gblock, 0, stream>>>(
        y0, scale1, shift1, w1, b1, x, out);
}